// LightGCN_59468117181010
// MI455X (gfx1250) — compile-verified
//
#include <hip/hip_runtime.h>
#include <hip/hip_bf16.h>
#include <stddef.h>

// ---------------------------------------------------------------------------
// LightGCN on gfx1250 (wave32).
//   1) zero counts; histogram edge_rows
//   2) exclusive scan (block scan -> block-sum scan -> add offsets) => row_ptr
//   3) zero fill; reorder packed {col, val} into CSR order (int2, 8B/edge)
//   4) concat [user_emb; item_emb] -> x
//   5) gather_init: out = all_emb[batch indices]
//   6) 3x { spmm_csr (wave/row, lane = float4 of dims, NO atomics;
//            edge stream staged via global_load_async_to_lds_b64 + ASYNCcnt,
//            broadcast back via uniform ds_load_b64);
//           gather_add into out (last layer folds the /4) ; swap x/y }
// x (61.4MB) is L2-resident (192MB L2) so the per-edge row gathers hit L2.
// ---------------------------------------------------------------------------

#define EMB 128          // floats per row
#define EMB4 32          // float4 per row
#define TBSZ 256         // threads per block (8 wave32)
#define WPB 8            // waves per block
#define SCAN_T 256
#define SCAN_IPT 8
#define SCAN_CHUNK (SCAN_T * SCAN_IPT)   // 2048

// ---- gfx1250 async VMEM->LDS path (guarded; fallback = v_readlane path) ----
#if defined(__has_builtin)
#  if __has_builtin(__builtin_amdgcn_global_load_async_to_lds_b64)
#    define HAVE_ASYNC_LDS 1
#  endif
#  if __has_builtin(__builtin_amdgcn_s_wait_asynccnt)
#    define HAVE_WAIT_ASYNC 1
#  endif
#endif

#if defined(HAVE_ASYNC_LDS)
typedef int v2i __attribute__((ext_vector_type(2)));
// Pointer-to-AS1-v2i and pointer-to-AS3-v2i (exact pointee types the builtin
// expects, per the compiler's own diagnostic).
typedef __attribute__((address_space(1))) v2i* gptr_v2i;
typedef __attribute__((address_space(3))) v2i* lptr_v2i;

// s_wait_asynccnt intrinsics are commonly modeled IntrNoMem+HasSideEffects,
// which would allow the optimizer to hoist LDS reads above the wait even
// though the async engine's LDS write is only ordered by ASYNCcnt. Bracket
// with zero-cost compiler memory fences so no LDS access crosses the wait.
static __device__ __forceinline__ void wait_async0() {
  asm volatile("" ::: "memory");
#if defined(HAVE_WAIT_ASYNC)
  __builtin_amdgcn_s_wait_asynccnt(0);
#else
  asm volatile("s_wait_asynccnt 0" ::: "memory");
#endif
  asm volatile("" ::: "memory");
}
#endif

__global__ void k_zero_i32(int* __restrict__ p, int n) {
  int i = blockIdx.x * blockDim.x + threadIdx.x;
  if (i < n) p[i] = 0;
}

__global__ void k_hist(const int* __restrict__ rows, int* __restrict__ cnt, int nE) {
  int i = blockIdx.x * blockDim.x + threadIdx.x;
  if (i < nE) atomicAdd(&cnt[rows[i]], 1);
}

// Block-local exclusive scan of `cnt` into `rp`; per-block total into bsum.
__global__ void k_scan1(const int* __restrict__ cnt, int* __restrict__ rp,
                        int* __restrict__ bsum, int n) {
  __shared__ int s[SCAN_T];
  const int t = threadIdx.x;
  const int base = blockIdx.x * SCAN_CHUNK + t * SCAN_IPT;
  int loc[SCAN_IPT];
  int tsum = 0;
#pragma unroll
  for (int i = 0; i < SCAN_IPT; ++i) {
    int idx = base + i;
    int v = (idx < n) ? cnt[idx] : 0;
    loc[i] = v;
    tsum += v;
  }
  s[t] = tsum;
  __syncthreads();
  for (int off = 1; off < SCAN_T; off <<= 1) {
    int v = (t >= off) ? s[t - off] : 0;
    __syncthreads();
    if (t >= off) s[t] += v;
    __syncthreads();
  }
  int excl = s[t] - tsum;
  if (t == SCAN_T - 1) bsum[blockIdx.x] = s[SCAN_T - 1];
  int run = excl;
#pragma unroll
  for (int i = 0; i < SCAN_IPT; ++i) {
    int idx = base + i;
    if (idx < n) rp[idx] = run;
    run += loc[i];
  }
}

// Single-block exclusive scan of block sums (nb <= SCAN_T).
__global__ void k_scan2(const int* __restrict__ bsum, int* __restrict__ boff, int nb) {
  __shared__ int s[SCAN_T];
  const int t = threadIdx.x;
  int v = (t < nb) ? bsum[t] : 0;
  s[t] = v;
  __syncthreads();
  for (int off = 1; off < SCAN_T; off <<= 1) {
    int u = (t >= off) ? s[t - off] : 0;
    __syncthreads();
    if (t >= off) s[t] += u;
    __syncthreads();
  }
  if (t < nb) boff[t] = s[t] - v;
}

__global__ void k_scan3(int* __restrict__ rp, const int* __restrict__ boff, int n) {
  int i = blockIdx.x * blockDim.x + threadIdx.x;
  if (i < n) rp[i] += boff[i / SCAN_CHUNK];
}

// Pack {col, val_bits} into one int2 per edge in CSR order.
__global__ void k_reorder(const int* __restrict__ rows, const int* __restrict__ cols,
                          const float* __restrict__ vals, const int* __restrict__ rp,
                          int* __restrict__ fill, int2* __restrict__ epack, int nE) {
  int i = blockIdx.x * blockDim.x + threadIdx.x;
  if (i < nE) {
    int r = rows[i];
    int p = rp[r] + atomicAdd(&fill[r], 1);
    int2 pk;
    pk.x = cols[i];
    pk.y = __float_as_int(vals[i]);
    epack[p] = pk;
  }
}

__global__ void k_concat(const float4* __restrict__ ue, const float4* __restrict__ ie,
                         float4* __restrict__ x, int n4u, int n4i) {
  int i = blockIdx.x * blockDim.x + threadIdx.x;
  if (i < n4u) {
    x[i] = ue[i];
  } else if (i < n4u + n4i) {
    x[i] = ie[i - n4u];
  }
}

// One wave32 per node-row; lane owns a float4 (32 lanes x 4 = 128 dims).
// Edge stream: chunks of 32 edges. Async path stages the packed {col,val}
// stream straight into a private per-wave LDS slot (global_load_async_to_lds,
// ASYNCcnt ordering, no VGPR transit, no barriers), then broadcasts each
// edge via a uniform-address ds_load_b64. Fallback path uses v_readlane.
__global__ void k_spmm_csr(const float4* __restrict__ x4, float4* __restrict__ y4,
                           const int* __restrict__ rp, const int2* __restrict__ epack,
                           int nn, int nE) {
#if defined(HAVE_ASYNC_LDS)
  __shared__ int2 s_cv[WPB * 32];
  const int wbase = (threadIdx.x >> 5) << 5;
#endif
  const int lane = threadIdx.x & 31;
  int row = (int)(blockIdx.x * (blockDim.x >> 5) + (threadIdx.x >> 5));
  row = __builtin_amdgcn_readfirstlane(row);   // wave-uniform -> scalar regs
  if (row >= nn) return;

  const int start = __builtin_amdgcn_readfirstlane(rp[row]);
  const int end   = __builtin_amdgcn_readfirstlane((row + 1 < nn) ? rp[row + 1] : nE);

  float4 acc; acc.x = 0.f; acc.y = 0.f; acc.z = 0.f; acc.w = 0.f;

  for (int base = start; base < end; base += 32) {
    const int j = base + lane;
    int m = end - base;
    if (m > 32) m = 32;

#if defined(HAVE_ASYNC_LDS)
    if (j < end) {
      __builtin_amdgcn_global_load_async_to_lds_b64(
          (gptr_v2i)(epack + j), (lptr_v2i)(&s_cv[wbase + lane]), 0, 0);
    }
    wait_async0();
    if (j < end) {
      // Warm one line of this lane's gather target (global_prefetch_b8).
      int2 mycv = s_cv[wbase + lane];
      __builtin_prefetch((const void*)(x4 + (size_t)mycv.x * EMB4 + lane), 0, 3);
    }
    for (int t = 0; t < m; ++t) {
      const int2 cv = s_cv[wbase + t];          // uniform addr -> LDS broadcast
      const int cc = cv.x;
      const float vv = __int_as_float(cv.y);
      const float4 xv = x4[(size_t)cc * EMB4 + lane];
      acc.x += vv * xv.x;
      acc.y += vv * xv.y;
      acc.z += vv * xv.z;
      acc.w += vv * xv.w;
    }
#else
    int2 cv_l; cv_l.x = 0; cv_l.y = 0;
    if (j < end) {
      cv_l = epack[j];
      __builtin_prefetch((const void*)(x4 + (size_t)cv_l.x * EMB4 + lane), 0, 3);
    }
    for (int t = 0; t < m; ++t) {
      const int cc = __builtin_amdgcn_readlane(cv_l.x, t);
      const float vv = __int_as_float(__builtin_amdgcn_readlane(cv_l.y, t));
      const float4 xv = x4[(size_t)cc * EMB4 + lane];
      acc.x += vv * xv.x;
      acc.y += vv * xv.y;
      acc.z += vv * xv.z;
      acc.w += vv * xv.w;
    }
#endif
  }
  y4[(size_t)row * EMB4 + lane] = acc;
}

// out[0 .. batch*EMB) = user rows ; out[batch*EMB .. 2*batch*EMB) = item rows
__global__ void k_gather_init(const float4* __restrict__ ue, const float4* __restrict__ ie,
                              const int* __restrict__ users, const int* __restrict__ items,
                              float4* __restrict__ out, int batch) {
  int i = blockIdx.x * blockDim.x + threadIdx.x;
  int n = batch * EMB4;
  if (i < n) {
    int b = i >> 5, q = i & 31;
    out[i] = ue[(size_t)users[b] * EMB4 + q];
    out[(size_t)n + i] = ie[(size_t)items[b] * EMB4 + q];
  }
}

__global__ void k_gather_add(const float4* __restrict__ src,
                             const int* __restrict__ users, const int* __restrict__ items,
                             float4* __restrict__ out, int batch, int nUsers, float scale) {
  int i = blockIdx.x * blockDim.x + threadIdx.x;
  int n = batch * EMB4;
  if (i < n) {
    int b = i >> 5, q = i & 31;
    float4 o = out[i];
    float4 s0 = src[(size_t)users[b] * EMB4 + q];
    o.x = (o.x + s0.x) * scale;
    o.y = (o.y + s0.y) * scale;
    o.z = (o.z + s0.z) * scale;
    o.w = (o.w + s0.w) * scale;
    out[i] = o;

    size_t ii = (size_t)n + i;
    float4 o2 = out[ii];
    float4 s1 = src[(size_t)(nUsers + items[b]) * EMB4 + q];
    o2.x = (o2.x + s1.x) * scale;
    o2.y = (o2.y + s1.y) * scale;
    o2.z = (o2.z + s1.z) * scale;
    o2.w = (o2.w + s1.w) * scale;
    out[ii] = o2;
  }
}

static inline int cdiv(long long a, long long b) { return (int)((a + b - 1) / b); }

extern "C" void kernel_launch(void* const* d_in, const int* in_sizes, int n_in,
                              void* d_out, int out_size, void* d_ws, size_t ws_size,
                              hipStream_t stream) {
  const float* ue    = (const float*)d_in[0];
  const float* ie    = (const float*)d_in[1];
  const int*   erow  = (const int*)d_in[2];
  const int*   ecolI = (const int*)d_in[3];
  const float* evalI = (const float*)d_in[4];
  const int*   users = (const int*)d_in[5];
  const int*   items = (const int*)d_in[6];

  const int nU    = in_sizes[0] / EMB;
  const int nI    = in_sizes[1] / EMB;
  const int nE    = in_sizes[2];
  const int batch = in_sizes[5];
  const int nn    = nU + nI;

  // ---- workspace layout (all 16B aligned by construction) ----
  const size_t nodef = (size_t)nn * EMB;
  float* x     = (float*)d_ws;                 // nn*128 f32
  float* y     = x + nodef;                    // nn*128 f32
  int*   rp    = (int*)(y + nodef);            // nn
  int*   cnt   = rp + nn;                      // nn (histogram, then fill)
  int*   bsum  = cnt + nn;                     // <=256
  int*   boff  = bsum + 256;                   // <=256
  int2*  epack = (int2*)(boff + 256);          // nE packed {col, val_bits}

  const int TB = TBSZ;
  const int nblkScan = cdiv(nn, SCAN_CHUNK);   // 59 for nn=120000 (<=256)

  // 1) CSR build
  k_zero_i32<<<cdiv(nn, TB), TB, 0, stream>>>(cnt, nn);
  k_hist<<<cdiv(nE, TB), TB, 0, stream>>>(erow, cnt, nE);
  k_scan1<<<nblkScan, SCAN_T, 0, stream>>>(cnt, rp, bsum, nn);
  k_scan2<<<1, SCAN_T, 0, stream>>>(bsum, boff, nblkScan);
  k_scan3<<<cdiv(nn, TB), TB, 0, stream>>>(rp, boff, nn);
  k_zero_i32<<<cdiv(nn, TB), TB, 0, stream>>>(cnt, nn);
  k_reorder<<<cdiv(nE, TB), TB, 0, stream>>>(erow, ecolI, evalI, rp, cnt, epack, nE);

  // 2) x = concat(user_emb, item_emb)
  const int n4u = nU * EMB4, n4i = nI * EMB4;
  k_concat<<<cdiv((long long)n4u + n4i, TB), TB, 0, stream>>>(
      (const float4*)ue, (const float4*)ie, (float4*)x, n4u, n4i);

  // 3) out = layer-0 contribution
  const int gthreads = batch * EMB4;
  k_gather_init<<<cdiv(gthreads, TB), TB, 0, stream>>>(
      (const float4*)ue, (const float4*)ie, users, items, (float4*)d_out, batch);

  // 4) 3 propagation layers, ping-pong x<->y; accumulate gathered rows into out
  const int spmmBlocks = cdiv((long long)nn * 32, TB);  // one wave per row
  float* cur = x;
  float* nxt = y;
  for (int layer = 0; layer < 3; ++layer) {
    k_spmm_csr<<<spmmBlocks, TB, 0, stream>>>(
        (const float4*)cur, (float4*)nxt, rp, epack, nn, nE);
    const float scale = (layer == 2) ? 0.25f : 1.0f;   // fold /(L+1) into last pass
    k_gather_add<<<cdiv(gthreads, TB), TB, 0, stream>>>(
        (const float4*)nxt, users, items, (float4*)d_out, batch, nU, scale);
    float* t = cur; cur = nxt; nxt = t;
  }

  (void)n_in; (void)out_size; (void)ws_size;
}